// ACTDistillM2LossHead_31705448579452
// MI455X (gfx1250) — compile-verified
//
#include <hip/hip_runtime.h>

// ---------------------------------------------------------------------------
// ACT distill M2 loss, MI455X (gfx1250, wave32).
// HBM-bound: 2 streamed passes over student+teacher logits (~2.1 GB => ~90us
// floor @ 23.3 TB/s). Block reductions use v_wmma_f32_16x16x4_f32 (B = ones)
// as a cross-lane reducer. Fully deterministic (no float atomics).
// ---------------------------------------------------------------------------

#define BB   4
#define TT   1024
#define TM   1023          // shifted time length
#define VV   32000
#define V4   8000          // VV / 4 (float4 chunks)
#define ROWS (BB * TM)     // 4092
#define IGN  (-100)

#define L2E 1.4426950408889634f
#define LN2 0.6931471805599453f

typedef float v2f __attribute__((ext_vector_type(2)));
typedef float v8f __attribute__((ext_vector_type(8)));

__device__ __forceinline__ float ex2(float x) { return __builtin_amdgcn_exp2f(x); }
__device__ __forceinline__ float lg2(float x) { return __builtin_amdgcn_logf(x); }

// ---- WMMA cross-lane reduction --------------------------------------------
// A (16x4 f32): lane m, VGPR0 -> A[m][0] (lanes 0-15) / A[m][2] (lanes 16-31);
// VGPR1 = 0. B = ones.  D[m][n] = x[m] + x[m+16].
// Lane L accumulates c[0..7] = sum over M in {0..7}+8*(L/16); one shfl_xor(16)
// then yields the full 32-lane sum in every lane.  EXEC must be all ones.
__device__ __forceinline__ float waveSumWMMA(float x) {
  v2f a; a[0] = x;   a[1] = 0.0f;
  v2f b; b[0] = 1.0f; b[1] = 1.0f;
  v8f c = {};
  c = __builtin_amdgcn_wmma_f32_16x16x4_f32(false, a, false, b, (short)0, c, false, false);
  float s = c[0] + c[1] + c[2] + c[3] + c[4] + c[5] + c[6] + c[7];
  s += __shfl_xor(s, 16, 32);
  return s;
}

// Deterministic 256-thread block sum (8 waves). All threads return the total.
__device__ __forceinline__ float blockSum256(float x, float* lds) {
  float w = waveSumWMMA(x);
  int lane = threadIdx.x & 31, wid = threadIdx.x >> 5;
  __syncthreads();                 // protect lds reuse across calls
  if (lane == 0) lds[wid] = w;
  __syncthreads();
  float r = lds[0];
#pragma unroll
  for (int i = 1; i < 8; ++i) r += lds[i];
  return r;
}

// Combine two online-softmax states (m1,s1) <- (m1,s1) + (m2,s2)
__device__ __forceinline__ void lseCombine(float& m1, float& s1, float m2, float s2) {
  float M = fmaxf(m1, m2);
  s1 = s1 * ex2((m1 - M) * L2E) + s2 * ex2((m2 - M) * L2E);
  m1 = M;
}

// ---- Pass 1: per-row logsumexp (student & teacher) + label log-probs ------
__global__ void k_pass1(const float* __restrict__ stu, const float* __restrict__ tea,
                        const long long* __restrict__ lab,
                        float* __restrict__ lse_s, float* __restrict__ lse_t,
                        float* __restrict__ pts, float* __restrict__ ptt,
                        float* __restrict__ msk) {
  int r = blockIdx.x;
  int b = r / TM, t = r % TM;
  size_t base = ((size_t)b * TT + t) * VV;
  const float4* ps = (const float4*)(stu + base);
  const float4* pt = (const float4*)(tea + base);

  float ms = -3.0e38f, ss = 0.0f;
  float mt = -3.0e38f, st = 0.0f;

  for (int i = threadIdx.x; i < V4; i += 256) {
    float4 xs = ps[i];
    float4 xt = pt[i];
    float m4 = fmaxf(fmaxf(xs.x, xs.y), fmaxf(xs.z, xs.w));
    float s4 = ex2((xs.x - m4) * L2E) + ex2((xs.y - m4) * L2E) +
               ex2((xs.z - m4) * L2E) + ex2((xs.w - m4) * L2E);
    lseCombine(ms, ss, m4, s4);
    m4 = fmaxf(fmaxf(xt.x, xt.y), fmaxf(xt.z, xt.w));
    s4 = ex2((xt.x - m4) * L2E) + ex2((xt.y - m4) * L2E) +
         ex2((xt.z - m4) * L2E) + ex2((xt.w - m4) * L2E);
    lseCombine(mt, st, m4, s4);
  }

  // wave32 butterfly combine of (m,s) states
#pragma unroll
  for (int off = 16; off; off >>= 1) {
    float mo = __shfl_xor(ms, off, 32);
    float so = __shfl_xor(ss, off, 32);
    lseCombine(ms, ss, mo, so);
    mo = __shfl_xor(mt, off, 32);
    so = __shfl_xor(st, off, 32);
    lseCombine(mt, st, mo, so);
  }

  __shared__ float lm[8][4];
  int lane = threadIdx.x & 31, wid = threadIdx.x >> 5;
  if (lane == 0) { lm[wid][0] = ms; lm[wid][1] = ss; lm[wid][2] = mt; lm[wid][3] = st; }
  __syncthreads();

  if (threadIdx.x == 0) {
    float Ms = lm[0][0], Ss = lm[0][1], Mt = lm[0][2], St = lm[0][3];
#pragma unroll
    for (int i = 1; i < 8; ++i) {
      lseCombine(Ms, Ss, lm[i][0], lm[i][1]);
      lseCombine(Mt, St, lm[i][2], lm[i][3]);
    }
    float Ls = Ms + lg2(Ss) * LN2;
    float Lt = Mt + lg2(St) * LN2;
    lse_s[r] = Ls;
    lse_t[r] = Lt;

    long long l = lab[(size_t)b * TT + t + 1];
    float m = (l != IGN) ? 1.0f : 0.0f;
    int idx = (int)l;
    idx = idx < 0 ? 0 : (idx > VV - 1 ? VV - 1 : idx);
    pts[r] = m * (stu[base + idx] - Ls);
    ptt[r] = m * (tea[base + idx] - Lt);
    msk[r] = m;
  }
}

// ---- Per-sequence masked sums ---------------------------------------------
__global__ void k_seqstats(const float* __restrict__ pts, const float* __restrict__ ptt,
                           const float* __restrict__ msk, float* __restrict__ stats) {
  __shared__ float lds[8];
  int b = blockIdx.x;
  float a = 0.0f, c = 0.0f, n = 0.0f;
  for (int t = threadIdx.x; t < TM; t += 256) {
    int r = b * TM + t;
    a += pts[r]; c += ptt[r]; n += msk[r];
  }
  float ra = blockSum256(a, lds);
  float rc = blockSum256(c, lds);
  float rn = blockSum256(n, lds);
  if (threadIdx.x == 0) { stats[b] = ra; stats[4 + b] = rc; stats[8 + b] = rn; }
}

// ---- Alphas -> per-sequence mix coefficients ------------------------------
// coef[b]   = term added to teacher log-prob inside logaddexp
// coef[4+b] = term added to student log-prob inside logaddexp
__global__ void k_alpha(const float* __restrict__ stats, float* __restrict__ coef) {
  if (threadIdx.x == 0 && blockIdx.x == 0) {
    float lps[4], lpt[4];
#pragma unroll
    for (int b = 0; b < 4; ++b) {
      float n = fmaxf(stats[8 + b], 1.0f);
      lps[b] = stats[b] / n;
      lpt[b] = stats[4 + b] / n;
    }
#pragma unroll
    for (int i = 0; i < 2; ++i) {
      int pb = 2 * i, nb = 2 * i + 1;
      float r1 = ex2((lpt[pb] - lps[pb]) * L2E);              // exp(logp_t - logp_s), pos
      float a1 = 1.0f - (1.0f - 0.1f) / (r1 + 1e-5f);
      a1 = fminf(fmaxf(a1, 0.01f), 0.1f);
      float r2 = ex2((lps[nb] - lpt[nb]) * L2E);              // exp(logp_s - logp_t), neg
      float a2 = 1.0f - (1.0f - 0.1f) / (r2 + 1e-5f);
      a2 = fminf(fmaxf(a2, 0.01f), 0.1f);
      coef[pb]     = lg2(a1) * LN2;          // log(alpha1) on teacher
      coef[4 + pb] = lg2(1.0f - a1) * LN2;   // log1p(-alpha1) on student
      coef[nb]     = lg2(1.0f - a2) * LN2;   // log1p(-alpha2) on teacher
      coef[4 + nb] = lg2(a2) * LN2;          // log(alpha2) on student
    }
  }
}

// ---- Pass 2: mix-KL per row ------------------------------------------------
__device__ __forceinline__ float klTerm(float xs, float xt, float Ls, float Lt,
                                        float cT, float cS, bool pos) {
  float sl = xs - Ls;
  float tl = xt - Lt;
  float u = cT + tl;
  float v = cS + sl;
  float hi = fmaxf(u, v), lo = fminf(u, v);
  float mix = hi + lg2(1.0f + ex2((lo - hi) * L2E)) * LN2;   // logaddexp(u, v)
  float p = pos ? tl : sl;
  return ex2(p * L2E) * (p - mix);
}

__global__ void k_pass2(const float* __restrict__ stu, const float* __restrict__ tea,
                        const float* __restrict__ lse_s, const float* __restrict__ lse_t,
                        const float* __restrict__ msk, const float* __restrict__ coef,
                        float* __restrict__ kl) {
  int r = blockIdx.x;
  int b = r / TM, t = r % TM;
  size_t base = ((size_t)b * TT + t) * VV;
  const float4* ps = (const float4*)(stu + base);
  const float4* pt = (const float4*)(tea + base);
  float Ls = lse_s[r], Lt = lse_t[r];
  float cT = coef[b], cS = coef[4 + b];
  bool pos = ((b & 1) == 0);

  float acc = 0.0f;
  for (int i = threadIdx.x; i < V4; i += 256) {
    float4 xs = ps[i];
    float4 xt = pt[i];
    acc += klTerm(xs.x, xt.x, Ls, Lt, cT, cS, pos);
    acc += klTerm(xs.y, xt.y, Ls, Lt, cT, cS, pos);
    acc += klTerm(xs.z, xt.z, Ls, Lt, cT, cS, pos);
    acc += klTerm(xs.w, xt.w, Ls, Lt, cT, cS, pos);
  }

  __shared__ float lds[8];
  float s = blockSum256(acc, lds);
  if (threadIdx.x == 0) kl[r] = s * msk[r];
}

// ---- Per-sequence masked mean of KL ---------------------------------------
__global__ void k_seqkl(const float* __restrict__ kl, const float* __restrict__ stats,
                        float* __restrict__ klseq) {
  __shared__ float lds[8];
  int b = blockIdx.x;
  float a = 0.0f;
  for (int t = threadIdx.x; t < TM; t += 256) a += kl[b * TM + t];
  float s = blockSum256(a, lds);
  if (threadIdx.x == 0) klseq[b] = s / fmaxf(stats[8 + b], 1.0f);
}

// ---- Final scalar ----------------------------------------------------------
__global__ void k_out(const float* __restrict__ klseq, float* __restrict__ out) {
  if (threadIdx.x == 0 && blockIdx.x == 0) {
    const float BETA = 1.0f;
    float pos = 0.5f * (klseq[0] + klseq[2]);
    float neg = 0.5f * (klseq[1] + klseq[3]);
    out[0] = (2.0f - BETA) * pos + BETA * neg;
  }
}

// ---- Host launch -----------------------------------------------------------
extern "C" void kernel_launch(void* const* d_in, const int* in_sizes, int n_in,
                              void* d_out, int out_size, void* d_ws, size_t ws_size,
                              hipStream_t stream) {
  const float* stu = (const float*)d_in[0];
  const float* tea = (const float*)d_in[1];
  const long long* lab = (const long long*)d_in[2];
  float* out = (float*)d_out;

  // Workspace layout (floats), all fully written before read each call.
  float* ws    = (float*)d_ws;
  float* lse_s = ws + 0;          // ROWS
  float* lse_t = ws + 4096;       // ROWS
  float* pts   = ws + 8192;       // ROWS
  float* ptt   = ws + 12288;      // ROWS
  float* msk   = ws + 16384;      // ROWS
  float* kl    = ws + 20480;      // ROWS
  float* stats = ws + 24576;      // 12: sum_pts[4], sum_ptt[4], cnt[4]
  float* coef  = ws + 24592;      // 8: teacher coef[4], student coef[4]
  float* klseq = ws + 24604;      // 4

  k_pass1<<<ROWS, 256, 0, stream>>>(stu, tea, lab, lse_s, lse_t, pts, ptt, msk);
  k_seqstats<<<BB, 256, 0, stream>>>(pts, ptt, msk, stats);
  k_alpha<<<1, 32, 0, stream>>>(stats, coef);
  k_pass2<<<ROWS, 256, 0, stream>>>(stu, tea, lse_s, lse_t, msk, coef, kl);
  k_seqkl<<<BB, 256, 0, stream>>>(kl, stats, klseq);
  k_out<<<1, 32, 0, stream>>>(klseq, out);
  (void)in_sizes; (void)n_in; (void)out_size; (void)ws_size;
}